// Preprocessor_13297218748833
// MI455X (gfx1250) — compile-verified
//
#include <hip/hip_runtime.h>

typedef __attribute__((ext_vector_type(2))) float v2f;
typedef __attribute__((ext_vector_type(8))) float v8f;
typedef __attribute__((ext_vector_type(4))) float f4;
typedef __attribute__((ext_vector_type(4))) int   v4i;

// Problem constants
static constexpr int IMG_H = 4096;
static constexpr int IMG_W = 4096;
static constexpr int NCH   = 4;
static constexpr int TGRID = 8;
static constexpr int TILE  = 512;              // IMG_H / TGRID
static constexpr int NBINS = 256;
static constexpr int AREA  = TILE * TILE;      // 262144
static constexpr int CLIPV = 40960;            // max(int(40*AREA/256),1)

// ---- CDNA5 async-to-LDS path (guarded: falls back if builtins absent) ----
#if defined(__has_builtin)
#if __has_builtin(__builtin_amdgcn_global_load_async_to_lds_b128)
#define HAS_ASYNC_LDS 1
#endif
#endif
#ifndef HAS_ASYNC_LDS
#define HAS_ASYNC_LDS 0
#endif

#if HAS_ASYNC_LDS
// Builtin signature (from clang diagnostic): (v4i AS1*, v4i AS3*, imm, imm)
typedef __attribute__((address_space(1))) v4i v4i_g;
typedef __attribute__((address_space(3))) v4i v4i_l;
// copy 16B/lane from global to LDS, tracked by ASYNCcnt (no VGPR data)
#define ASYNC_CP16(gp, lp)                                                   \
  __builtin_amdgcn_global_load_async_to_lds_b128((v4i_g*)(gp), (v4i_l*)(lp), \
                                                 0, 0)
#if __has_builtin(__builtin_amdgcn_s_wait_asynccnt)
#define WAIT_ASYNC(n) __builtin_amdgcn_s_wait_asynccnt(n)
#else
#define WAIT_ASYNC(n) asm volatile("s_wait_asynccnt %0" ::"n"(n) : "memory")
#endif
#endif

// ---- helpers -------------------------------------------------------------

// Monotonic float<->uint mapping so unsigned atomicMin/Max order floats.
__device__ __forceinline__ unsigned fmap(float f) {
  unsigned b = __float_as_uint(f);
  return (b & 0x80000000u) ? ~b : (b | 0x80000000u);
}
__device__ __forceinline__ float funmap(unsigned u) {
  unsigned b = (u & 0x80000000u) ? (u ^ 0x80000000u) : ~u;
  return __uint_as_float(b);
}

// vi = trunc(((x-mn)/d) * 255) clamped to [0,255]  (matches astype(uint8) trunc)
__device__ __forceinline__ int quant(float f, float mn, float d) {
  float xn = (f - mn) / d;
  int q = (int)(xn * 255.0f);
  return q < 0 ? 0 : (q > 255 ? 255 : q);
}

// ---- kernel 0: reset workspace ------------------------------------------
__global__ void k_init(unsigned* __restrict__ mm, unsigned* __restrict__ hist) {
  int gid = blockIdx.x * blockDim.x + threadIdx.x;
  for (int i = gid; i < NCH * TGRID * TGRID * NBINS; i += gridDim.x * blockDim.x)
    hist[i] = 0u;
  if (gid < NCH) {
    mm[2 * gid + 0] = 0xFFFFFFFFu;  // min slot
    mm[2 * gid + 1] = 0x00000000u;  // max slot
  }
}

// ---- kernel 1: per-channel min/max --------------------------------------
static constexpr int MM_BLOCKS_PER_CH = 2048;   // 2048 blk * 256 thr * 8 f4 = 16M floats
__global__ void k_minmax(const float* __restrict__ x, unsigned* __restrict__ mm) {
  int ch = blockIdx.x / MM_BLOCKS_PER_CH;
  int b  = blockIdx.x % MM_BLOCKS_PER_CH;
  const f4* p = (const f4*)(x + (size_t)ch * IMG_H * (size_t)IMG_W);
  int base = b * 2048 + threadIdx.x;
  float mn = 3.4e38f, mx = -3.4e38f;
#pragma unroll
  for (int i = 0; i < 8; ++i) {
    f4 v = p[base + i * 256];
    mn = fminf(mn, fminf(fminf(v.x, v.y), fminf(v.z, v.w)));
    mx = fmaxf(mx, fmaxf(fmaxf(v.x, v.y), fmaxf(v.z, v.w)));
  }
#pragma unroll
  for (int m = 16; m >= 1; m >>= 1) {
    mn = fminf(mn, __shfl_xor(mn, m, 32));
    mx = fmaxf(mx, __shfl_xor(mx, m, 32));
  }
  if ((threadIdx.x & 31) == 0) {
    atomicMin(&mm[ch * 2 + 0], fmap(mn));
    atomicMax(&mm[ch * 2 + 1], fmap(mx));
  }
}

// ---- kernel 2: per-tile histograms --------------------------------------
// grid = NCH*64*4 blocks of 256 threads; each block: 128 rows x 512 cols.
// Async path: double-buffered GLOBAL_LOAD_ASYNC_TO_LDS_B128 staging of 8-row
// chunks (16KB), each wave32 issues and consumes only its own LDS region, so
// the only sync needed is the wave-local s_wait_asynccnt. Per-wave privatized
// LDS histograms (ds_add) kill atomic contention; 40KB LDS/block of 320KB WGP.
__global__ void k_hist(const float* __restrict__ x, const unsigned* __restrict__ mm,
                       unsigned* __restrict__ hist) {
  __shared__ unsigned lh[8 * NBINS];   // one 256-bin hist per wave32
#if HAS_ASYNC_LDS
  __shared__ f4 stage[2][1024];        // 2 x 16KB double buffer (8 rows x 512)
#endif
  int ch   = blockIdx.x >> 8;
  int rest = blockIdx.x & 255;
  int tile = rest >> 2;
  int sub  = rest & 3;
  int tyi = tile >> 3, txi = tile & 7;
  float mn = funmap(mm[ch * 2 + 0]);
  float mx = funmap(mm[ch * 2 + 1]);
  float d  = mx - mn;
  int t = threadIdx.x;
#pragma unroll
  for (int i = 0; i < 8; ++i) lh[t + i * 256] = 0u;
  __syncthreads();
  const float* base = x + (size_t)ch * IMG_H * (size_t)IMG_W +
                      (size_t)(tyi * TILE + sub * 128) * IMG_W + txi * TILE;
  int wave = t >> 5, lane = t & 31;
  unsigned* wh = lh + (wave << 8);

#if HAS_ASYNC_LDS
  // pre-issue chunks 0 and 1 (4 async b128 per wave per chunk)
#pragma unroll
  for (int pc = 0; pc < 2; ++pc) {
#pragma unroll
    for (int k = 0; k < 4; ++k) {
      int j   = wave * 128 + k * 32 + lane;     // f4 index within chunk
      int row = pc * 8 + (j >> 7);
      int col = (j & 127) << 2;
      ASYNC_CP16(base + (size_t)row * IMG_W + col, &stage[pc][j]);
    }
  }
  for (int c = 0; c < 16; ++c) {
    int b = c & 1;
    if (c < 15) { WAIT_ASYNC(4); } else { WAIT_ASYNC(0); }  // chunk c landed
#pragma unroll
    for (int k = 0; k < 4; ++k) {
      int j = wave * 128 + k * 32 + lane;
      f4 v = stage[b][j];                        // ds_load_b128
      atomicAdd(&wh[quant(v.x, mn, d)], 1u);
      atomicAdd(&wh[quant(v.y, mn, d)], 1u);
      atomicAdd(&wh[quant(v.z, mn, d)], 1u);
      atomicAdd(&wh[quant(v.w, mn, d)], 1u);
    }
    if (c + 2 < 16) {                            // refill consumed buffer
#pragma unroll
      for (int k = 0; k < 4; ++k) {
        int j   = wave * 128 + k * 32 + lane;
        int row = (c + 2) * 8 + (j >> 7);
        int col = (j & 127) << 2;
        ASYNC_CP16(base + (size_t)row * IMG_W + col, &stage[b][j]);
      }
    }
  }
#else
  for (int i = 0; i < 64; ++i) {
    int j   = t + i * 256;        // 0..16383 f4 in this 128x512 sub-tile
    int row = j >> 7;             // 128 f4 per row
    int c4  = (j & 127) << 2;
    const float* p = base + (size_t)row * IMG_W + c4;
    if (i + 4 < 64) {             // global_prefetch_b8 a few rows ahead
      int jn = t + (i + 4) * 256;
      __builtin_prefetch(base + (size_t)(jn >> 7) * IMG_W + ((jn & 127) << 2), 0, 0);
    }
    f4 v = *(const f4*)p;
    atomicAdd(&wh[quant(v.x, mn, d)], 1u);
    atomicAdd(&wh[quant(v.y, mn, d)], 1u);
    atomicAdd(&wh[quant(v.z, mn, d)], 1u);
    atomicAdd(&wh[quant(v.w, mn, d)], 1u);
  }
#endif
  __syncthreads();
  unsigned s = 0;
#pragma unroll
  for (int w = 0; w < 8; ++w) s += lh[w * 256 + t];
  atomicAdd(&hist[(ch * 64 + tile) * NBINS + t], s);
}

// ---- kernel 3: clip + redistribute + CDF via WMMA -----------------------
// One wave32 per (channel,tile). Inclusive scan of the 256-bin histogram as
// S = H(16x16) x U(16x16 upper-triangular ones), computed with four chained
// V_WMMA_F32_16X16X4_F32 (K split in chunks of 4), then a 16-row prefix fixup.
// Layouts per CDNA5 ISA 7.12.2:
//   A 16x4 f32 : lane<16 -> M=lane, {K0,K1}; lane>=16 -> M=lane-16, {K2,K3}
//   B 4x16 f32 : lane<16 -> N=lane, rows {K0,K1}; lane>=16 -> N=lane-16, {K2,K3}
//   C/D 16x16  : vgpr v, lane<16 -> D[v][lane]; lane>=16 -> D[v+8][lane-16]
__global__ void k_lut(const unsigned* __restrict__ hist, float* __restrict__ lut) {
  __shared__ float srt[16];   // row totals
  __shared__ float srp[16];   // exclusive row prefixes
  int id = blockIdx.x;        // ch*64 + tile
  const unsigned* hp = hist + id * NBINS;
  int lane = threadIdx.x;
  int r    = lane & 15;
  int koff = (lane < 16) ? 0 : 2;

  unsigned hraw[8];
#pragma unroll
  for (int j = 0; j < 4; ++j) {
    hraw[2 * j]     = hp[16 * r + 4 * j + koff];
    hraw[2 * j + 1] = hp[16 * r + 4 * j + koff + 1];
  }
  // total excess over clip (lanes jointly cover all 256 bins exactly once)
  float ex = 0.f;
#pragma unroll
  for (int e = 0; e < 8; ++e) ex += fmaxf((float)hraw[e] - (float)CLIPV, 0.f);
#pragma unroll
  for (int m = 16; m >= 1; m >>= 1) ex += __shfl_xor(ex, m, 32);
  float add = ex * (1.0f / 256.0f);

  v2f a[4], b[4];
  int n = lane & 15;
#pragma unroll
  for (int j = 0; j < 4; ++j) {
    a[j].x = fminf((float)hraw[2 * j],     (float)CLIPV) + add;
    a[j].y = fminf((float)hraw[2 * j + 1], (float)CLIPV) + add;
    int k0 = 4 * j + koff;
    b[j].x = (k0     <= n) ? 1.0f : 0.0f;   // U[k][n] = (k <= n)
    b[j].y = (k0 + 1 <= n) ? 1.0f : 0.0f;
  }

  v8f acc = {};
  acc = __builtin_amdgcn_wmma_f32_16x16x4_f32(false, a[0], false, b[0], (short)0, acc, false, false);
  acc = __builtin_amdgcn_wmma_f32_16x16x4_f32(false, a[1], false, b[1], (short)0, acc, false, false);
  acc = __builtin_amdgcn_wmma_f32_16x16x4_f32(false, a[2], false, b[2], (short)0, acc, false, false);
  acc = __builtin_amdgcn_wmma_f32_16x16x4_f32(false, a[3], false, b[3], (short)0, acc, false, false);

  // row totals live in column N=15 (lanes 15 and 31)
  if (lane == 15 || lane == 31) {
    int off = (lane < 16) ? 0 : 8;
#pragma unroll
    for (int v = 0; v < 8; ++v) srt[off + v] = acc[v];
  }
  __syncthreads();
  if (lane < 16) {
    float p = 0.f;
    for (int q = 0; q < lane; ++q) p += srt[q];
    srp[lane] = p;
  }
  __syncthreads();

  int off = (lane < 16) ? 0 : 8;
  const float scale = 255.0f / (float)AREA;
  float* lp = lut + id * NBINS;
#pragma unroll
  for (int v = 0; v < 8; ++v) {
    int M = v + off;
    float val = __builtin_rintf((acc[v] + srp[M]) * scale);  // jnp.round = RNE
    val = fminf(fmaxf(val, 0.f), 255.0f);
    lp[16 * M + n] = val;
  }
}

// ---- kernel 4: apply (bilinear blend of 4 tile LUTs) --------------------
__global__ void k_apply(const float* __restrict__ x, const unsigned* __restrict__ mm,
                        const float* __restrict__ lut, float* __restrict__ out) {
  size_t i  = (size_t)blockIdx.x * blockDim.x + threadIdx.x;  // f4 index
  int ch    = (int)(i >> 22);                                 // H*W/4 = 2^22
  int rem   = (int)(i & 4194303u);
  int y     = rem >> 10;                                      // 1024 f4 per row
  int x0    = (rem & 1023) << 2;
  float mn  = funmap(mm[ch * 2 + 0]);
  float mx  = funmap(mm[ch * 2 + 1]);
  float d   = mx - mn;

  float ty  = (float)y * (1.0f / 512.0f) - 0.5f;
  float tyf = floorf(ty);
  float ya  = ty - tyf;
  int ty1 = (int)tyf, ty2 = (int)tyf + 1;
  ty1 = ty1 < 0 ? 0 : (ty1 > 7 ? 7 : ty1);
  ty2 = ty2 < 0 ? 0 : (ty2 > 7 ? 7 : ty2);

  const float* lbase = lut + ch * 64 * NBINS;
  size_t poff = (size_t)ch * IMG_H * (size_t)IMG_W + (size_t)y * IMG_W + x0;
  f4 v = *(const f4*)(x + poff);

  float res[4];
#pragma unroll
  for (int p = 0; p < 4; ++p) {
    float f = (p == 0) ? v.x : (p == 1) ? v.y : (p == 2) ? v.z : v.w;
    int vi = quant(f, mn, d);
    float tx  = (float)(x0 + p) * (1.0f / 512.0f) - 0.5f;
    float txf = floorf(tx);
    float xa  = tx - txf;
    int tx1 = (int)txf, tx2 = (int)txf + 1;
    tx1 = tx1 < 0 ? 0 : (tx1 > 7 ? 7 : tx1);
    tx2 = tx2 < 0 ? 0 : (tx2 > 7 ? 7 : tx2);
    float l11 = lbase[((ty1 * 8 + tx1) << 8) + vi];
    float l12 = lbase[((ty1 * 8 + tx2) << 8) + vi];
    float l21 = lbase[((ty2 * 8 + tx1) << 8) + vi];
    float l22 = lbase[((ty2 * 8 + tx2) << 8) + vi];
    float rr  = (l11 * (1.f - xa) + l12 * xa) * (1.f - ya) +
                (l21 * (1.f - xa) + l22 * xa) * ya;
    float eq  = __builtin_rintf(rr) / 255.0f;
    res[p] = (vi == 0) ? 0.f : eq;
  }
  f4 o; o.x = res[0]; o.y = res[1]; o.z = res[2]; o.w = res[3];
  *(f4*)(out + poff) = o;
}

// ---- launch --------------------------------------------------------------
extern "C" void kernel_launch(void* const* d_in, const int* in_sizes, int n_in,
                              void* d_out, int out_size, void* d_ws, size_t ws_size,
                              hipStream_t stream) {
  const float* x = (const float*)d_in[0];
  float* out = (float*)d_out;

  // workspace layout: [0..63] minmax u32 (padded), then hist u32[65536], then lut f32[65536]
  unsigned* wsMM   = (unsigned*)d_ws;
  unsigned* wsHist = wsMM + 64;
  float*    wsLut  = (float*)(wsHist + NCH * TGRID * TGRID * NBINS);

  k_init  <<<64, 256, 0, stream>>>(wsMM, wsHist);
  k_minmax<<<NCH * MM_BLOCKS_PER_CH, 256, 0, stream>>>(x, wsMM);
  k_hist  <<<NCH * 64 * 4, 256, 0, stream>>>(x, wsMM, wsHist);
  k_lut   <<<NCH * 64, 32, 0, stream>>>(wsHist, wsLut);
  k_apply <<<(NCH * IMG_H * IMG_W / 4) / 256, 256, 0, stream>>>(x, wsMM, wsLut, out);
}